// GNN_69071664054740
// MI455X (gfx1250) — compile-verified
//
#include <hip/hip_runtime.h>

typedef float v2f __attribute__((ext_vector_type(2)));
typedef float v8f __attribute__((ext_vector_type(8)));

#define SELU_ALPHA 1.6732632423543772f
#define SELU_SCALE 1.0507009873554805f

__device__ __forceinline__ float selu_f(float x) {
  return x > 0.0f ? SELU_SCALE * x : SELU_SCALE * SELU_ALPHA * expm1f(x);
}

// ---------------- init ----------------
__global__ void k_node_init(float* deg, float* dsrc, int n) {
  int i = blockIdx.x * blockDim.x + threadIdx.x;
  if (i < n) { deg[i] = 1.0f; dsrc[i] = 0.0f; }  // self-loop weight 1 folded into deg
}

__global__ void k_small_init(float* pool, float* adjA, float* ssA, float* ssum,
                             float* ca, float* twom) {
  int i = blockIdx.x * blockDim.x + threadIdx.x;
  if (i < 1024) pool[i] = 0.f;
  if (i < 256) { adjA[i] = 0.f; ssA[i] = 0.f; }
  if (i < 16) { ssum[i] = 0.f; ca[i] = 0.f; }
  if (i == 0) *twom = 0.f;
}

// ---------------- degrees ----------------
__global__ void __launch_bounds__(256)
k_edge_deg(const int* __restrict__ src, const int* __restrict__ dst,
           const float* __restrict__ ew, float* deg, float* dsrc, float* twom, int E) {
  __shared__ float red[256];
  int gid = blockIdx.x * blockDim.x + threadIdx.x;
  float w = 0.f;
  if (gid < E) {
    w = ew[gid];
    atomicAdd(&deg[dst[gid]], w);
    atomicAdd(&dsrc[src[gid]], w);
  }
  red[threadIdx.x] = w;
  __syncthreads();
  for (int s = 128; s > 0; s >>= 1) {
    if (threadIdx.x < s) red[threadIdx.x] += red[threadIdx.x + s];
    __syncthreads();
  }
  if (threadIdx.x == 0) atomicAdd(twom, red[0]);
}

__global__ void k_dinv(const float* deg, float* dinv, int n) {
  int i = blockIdx.x * blockDim.x + threadIdx.x;
  if (i < n) dinv[i] = deg[i] > 0.f ? rsqrtf(deg[i]) : 0.f;
}

// ---------------- dense GEMM: C[nrows x 64] = A[nrows x 64] @ W[64 x 64] ----------------
// fp32 WMMA 16x16x4, one 16-row tile per wave, 4 column tiles, K loop of 16 steps.
__global__ void __launch_bounds__(128)
k_gemm_n64(const float* __restrict__ A, const float* __restrict__ W,
           float* __restrict__ C, int nrows) {
  __shared__ float sW[64 * 64];
  for (int i = threadIdx.x; i < 64 * 64; i += blockDim.x) sW[i] = W[i];
  __syncthreads();
  int lane = threadIdx.x & 31;
  int wave = threadIdx.x >> 5;
  int row0 = (blockIdx.x * 4 + wave) * 16;
  if (row0 >= nrows) return;               // wave-uniform; EXEC all-ones for WMMA
  int m = lane & 15;                       // A row / B,C col within tile
  int koff = (lane >> 4) << 1;             // K = vgpr + 2*(lane>=16)
  v8f acc[4] = {};
  for (int k0 = 0; k0 < 64; k0 += 4) {
    int kk = k0 + koff;
    const float* ap = A + (size_t)(row0 + m) * 64 + kk;
    v2f a; a.x = ap[0]; a.y = ap[1];
#pragma unroll
    for (int nt = 0; nt < 4; ++nt) {
      v2f b;
      b.x = sW[kk * 64 + nt * 16 + m];
      b.y = sW[(kk + 1) * 64 + nt * 16 + m];
      acc[nt] = __builtin_amdgcn_wmma_f32_16x16x4_f32(false, a, false, b,
                                                      (short)0, acc[nt], false, false);
    }
  }
  int rbase = row0 + ((lane >> 4) << 3);   // C row = j + 8*(lane>=16)
#pragma unroll
  for (int j = 0; j < 8; ++j) {
#pragma unroll
    for (int nt = 0; nt < 4; ++nt)
      C[(size_t)(rbase + j) * 64 + nt * 16 + m] = acc[nt][j];
  }
}

// ---------------- propagation ----------------
__global__ void k_prop_init(const float* __restrict__ tmp, const float* __restrict__ dinv,
                            float* __restrict__ agg, int total) {
  int i = blockIdx.x * blockDim.x + threadIdx.x;
  if (i < total) {
    float di = dinv[i >> 6];
    agg[i] = di * di * tmp[i];             // self-loop term: dinv^2 * (xW)
  }
}

__global__ void __launch_bounds__(256)
k_prop_edges(const int* __restrict__ src, const int* __restrict__ dst,
             const float* __restrict__ ew, const float* __restrict__ dinv,
             const float* __restrict__ tmp, float* __restrict__ agg, int E) {
  int lane = threadIdx.x & 31;
  int gw = (blockIdx.x * blockDim.x + threadIdx.x) >> 5;
  int nw = (gridDim.x * blockDim.x) >> 5;
  for (int e = gw; e < E; e += nw) {       // one edge per wave: 32 lanes x float2 = 64 feats
    int sn = src[e], dn = dst[e];
    float norm = dinv[sn] * ew[e] * dinv[dn];
    float2 hv = ((const float2*)(tmp + (size_t)sn * 64))[lane];
    float* dp = agg + (size_t)dn * 64 + lane * 2;
    atomicAdd(dp, norm * hv.x);
    atomicAdd(dp + 1, norm * hv.y);
  }
}

__global__ void k_selu_bias(const float* __restrict__ agg, const float* __restrict__ b,
                            float* __restrict__ h, int total) {
  int i = blockIdx.x * blockDim.x + threadIdx.x;
  if (i < total) h[i] = selu_f(agg[i] + b[i & 63]);
}

// ---------------- assignment: s = softmax(h @ Wp + bp), WMMA + half-wave softmax ----------------
__global__ void __launch_bounds__(128)
k_assign(const float* __restrict__ H, const float* __restrict__ Wp,
         const float* __restrict__ bp, float* __restrict__ S, int nrows) {
  __shared__ float sW[64 * 16];
  for (int i = threadIdx.x; i < 64 * 16; i += blockDim.x) sW[i] = Wp[i];
  __syncthreads();
  int lane = threadIdx.x & 31;
  int wave = threadIdx.x >> 5;
  int row0 = (blockIdx.x * 4 + wave) * 16;
  if (row0 >= nrows) return;
  int m = lane & 15;
  int koff = (lane >> 4) << 1;
  v8f acc = {};
  for (int k0 = 0; k0 < 64; k0 += 4) {
    int kk = k0 + koff;
    const float* ap = H + (size_t)(row0 + m) * 64 + kk;
    v2f a, b;
    a.x = ap[0]; a.y = ap[1];
    b.x = sW[kk * 16 + m];
    b.y = sW[(kk + 1) * 16 + m];
    acc = __builtin_amdgcn_wmma_f32_16x16x4_f32(false, a, false, b,
                                                (short)0, acc, false, false);
  }
  float bias = bp[m];
  int rbase = row0 + ((lane >> 4) << 3);
#pragma unroll
  for (int j = 0; j < 8; ++j) {
    float v = acc[j] + bias;               // row rbase+j lives across 16 lanes of half-wave
    float mx = v;
#pragma unroll
    for (int o = 1; o < 16; o <<= 1) mx = fmaxf(mx, __shfl_xor(mx, o, 32));
    float e = __expf(v - mx);
    float sm = e;
#pragma unroll
    for (int o = 1; o < 16; o <<= 1) sm += __shfl_xor(sm, o, 32);
    float val = e / sm;
    float s2 = val;                        // reference's s / s.sum() renormalization
#pragma unroll
    for (int o = 1; o < 16; o <<= 1) s2 += __shfl_xor(s2, o, 32);
    val /= s2;
    S[(size_t)(rbase + j) * 16 + m] = val;
  }
}

// ---------------- pooled features: pool[k][f] += s[n][k]*h[n][f] ----------------
__global__ void __launch_bounds__(256)
k_pool(const float* __restrict__ S, const float* __restrict__ H,
       float* __restrict__ pool, int n) {
  __shared__ float lp[1024];
  for (int i = threadIdx.x; i < 1024; i += blockDim.x) lp[i] = 0.f;
  __syncthreads();
  int lane = threadIdx.x & 31;
  int gw = (blockIdx.x * blockDim.x + threadIdx.x) >> 5;
  int nw = (gridDim.x * blockDim.x) >> 5;
  for (int node = gw; node < n; node += nw) {
    float2 hv = ((const float2*)(H + (size_t)node * 64))[lane];
    const float* sp = S + (size_t)node * 16;
#pragma unroll
    for (int k = 0; k < 16; ++k) {
      float sv = sp[k];
      atomicAdd(&lp[k * 64 + lane * 2], sv * hv.x);
      atomicAdd(&lp[k * 64 + lane * 2 + 1], sv * hv.y);
    }
  }
  __syncthreads();
  for (int i = threadIdx.x; i < 1024; i += blockDim.x) atomicAdd(&pool[i], lp[i]);
}

// ---------------- pooled adjacency: adj[k][c] += ew*s[src][k]*s[dst][c] ----------------
__global__ void __launch_bounds__(256)
k_adj(const int* __restrict__ src, const int* __restrict__ dst,
      const float* __restrict__ ew, const float* __restrict__ S,
      float* __restrict__ adjA, int E) {
  __shared__ float la[256];
  for (int i = threadIdx.x; i < 256; i += blockDim.x) la[i] = 0.f;
  __syncthreads();
  int lane = threadIdx.x & 31;
  int c = lane & 15;
  int kb = (lane >> 4) << 3;               // each lane covers 8 k's, fixed c
  int gw = (blockIdx.x * blockDim.x + threadIdx.x) >> 5;
  int nw = (gridDim.x * blockDim.x) >> 5;
  for (int e = gw; e < E; e += nw) {
    int sn = src[e], dn = dst[e];
    float sd = S[(size_t)dn * 16 + c] * ew[e];
    const float* sp = S + (size_t)sn * 16 + kb;
#pragma unroll
    for (int k = 0; k < 8; ++k)
      atomicAdd(&la[(kb + k) * 16 + c], sp[k] * sd);
  }
  __syncthreads();
  for (int i = threadIdx.x; i < 256; i += blockDim.x) atomicAdd(&adjA[i], la[i]);
}

// ---------------- s moments: ss = s^T s, ssum = sum s, ca = s^T d ----------------
__global__ void __launch_bounds__(256)
k_smoments(const float* __restrict__ S, const float* __restrict__ d,
           float* __restrict__ ssA, float* __restrict__ ssum, float* __restrict__ ca,
           int n) {
  __shared__ float lss[256];
  __shared__ float lsum[16];
  __shared__ float lca[16];
  for (int i = threadIdx.x; i < 256; i += blockDim.x) lss[i] = 0.f;
  if (threadIdx.x < 16) { lsum[threadIdx.x] = 0.f; lca[threadIdx.x] = 0.f; }
  __syncthreads();
  int lane = threadIdx.x & 31;
  int c = lane & 15;
  int kb = (lane >> 4) << 3;
  int gw = (blockIdx.x * blockDim.x + threadIdx.x) >> 5;
  int nw = (gridDim.x * blockDim.x) >> 5;
  for (int node = gw; node < n; node += nw) {
    const float* sp = S + (size_t)node * 16;
    float sc = sp[c];
    if (lane < 16) {
      atomicAdd(&lsum[lane], sc);
      atomicAdd(&lca[lane], sc * d[node]);
    }
#pragma unroll
    for (int k = 0; k < 8; ++k)
      atomicAdd(&lss[(kb + k) * 16 + c], sp[kb + k] * sc);
  }
  __syncthreads();
  for (int i = threadIdx.x; i < 256; i += blockDim.x) atomicAdd(&ssA[i], lss[i]);
  if (threadIdx.x < 16) {
    atomicAdd(&ssum[threadIdx.x], lsum[threadIdx.x]);
    atomicAdd(&ca[threadIdx.x], lca[threadIdx.x]);
  }
}

// ---------------- finalize: selu(pool), losses, normalized adjacency ----------------
__global__ void __launch_bounds__(256)
k_finalize(const float* __restrict__ pool, const float* __restrict__ adjA,
           const float* __restrict__ ssA, const float* __restrict__ ssum,
           const float* __restrict__ ca, const float* __restrict__ twom,
           float* __restrict__ out, int n, int outOff, int adjOff, int scalOff) {
  int t = threadIdx.x;
  for (int i = t; i < 1024; i += 256) out[outOff + i] = selu_f(pool[i]);
  __shared__ float sadj[256];
  __shared__ float srow[16];
  int r = t >> 4, c = t & 15;
  sadj[t] = (r == c) ? 0.f : adjA[t];      // zero diagonal
  __syncthreads();
  if (t < 16) {
    float rs = 0.f;
    for (int j = 0; j < 16; ++j) rs += sadj[t * 16 + j];
    srow[t] = sqrtf(rs) + 1e-15f;          // dd = sqrt(rowsum) + EPS
  }
  __syncthreads();
  out[adjOff + t] = sadj[t] / srow[r] / srow[c];
  if (t == 0) {
    float tr = 0.f, cc = 0.f;
    for (int k = 0; k < 16; ++k) { tr += adjA[k * 16 + k]; cc += ca[k] * ca[k]; }
    float tm = *twom;
    out[scalOff + 0] = -(tr - cc / tm) / tm;
  }
  if (t == 1) {
    float fn = 0.f;
    for (int i = 0; i < 256; ++i) fn += ssA[i] * ssA[i];
    fn = sqrtf(fn);
    float s = 0.f;
    for (int i = 0; i < 256; ++i) {
      float v = ssA[i] / fn - (((i >> 4) == (i & 15)) ? 0.25f : 0.f);  // I/sqrt(16)
      s += v * v;
    }
    out[scalOff + 1] = sqrtf(s);
  }
  if (t == 2) {
    float s = 0.f;
    for (int k = 0; k < 16; ++k) s += ssum[k] * ssum[k];
    out[scalOff + 2] = sqrtf(s) / (float)n * 4.0f - 1.0f;
  }
}

extern "C" void kernel_launch(void* const* d_in, const int* in_sizes, int n_in,
                              void* d_out, int out_size, void* d_ws, size_t ws_size,
                              hipStream_t stream) {
  const float* x  = (const float*)d_in[0];
  const int*   ei = (const int*)d_in[1];      // [2,E] int32 (JAX canonicalized)
  const float* ew = (const float*)d_in[2];
  const float* W1 = (const float*)d_in[3];
  const float* b1 = (const float*)d_in[4];
  const float* W2 = (const float*)d_in[5];
  const float* b2 = (const float*)d_in[6];
  const float* Wp = (const float*)d_in[7];
  const float* bp = (const float*)d_in[8];
  const int N = in_sizes[0] / 64;
  const int E = in_sizes[2];
  const int* src = ei;
  const int* dst = ei + E;
  float* out = (float*)d_out;

  // workspace carve-up
  float* ws = (float*)d_ws;
  float* tmp  = ws; ws += (size_t)N * 64;     // pre-propagation linear
  float* agg  = ws; ws += (size_t)N * 64;     // propagation accumulator
  float* h    = ws; ws += (size_t)N * 64;     // layer output (h1 then h2)
  float* deg  = ws; ws += N;
  float* dinv = ws; ws += N;
  float* dsrc = ws; ws += N;
  float* pool = ws; ws += 1024;
  float* adjA = ws; ws += 256;
  float* ssA  = ws; ws += 256;
  float* ssm  = ws; ws += 16;
  float* ca   = ws; ws += 16;
  float* twom = ws; ws += 1;

  float* s_out = out;                         // s written directly into d_out [N,16]
  const int outOff  = N * 16;
  const int adjOff  = outOff + 16 * 64;
  const int scalOff = adjOff + 256;
  (void)n_in; (void)out_size; (void)ws_size; (void)scalOff;

  const int total = N * 64;
  const int nbN = (N + 255) / 256;
  const int nbE = (E + 255) / 256;
  const int nbF = (total + 255) / 256;
  const int tiles = (N + 15) / 16;
  const int nbG = (tiles + 3) / 4;

  k_node_init<<<nbN, 256, 0, stream>>>(deg, dsrc, N);
  k_small_init<<<8, 256, 0, stream>>>(pool, adjA, ssA, ssm, ca, twom);
  k_edge_deg<<<nbE, 256, 0, stream>>>(src, dst, ew, deg, dsrc, twom, E);
  k_dinv<<<nbN, 256, 0, stream>>>(deg, dinv, N);

  // layer 1: h = selu(prop(x @ W1) + b1)
  k_gemm_n64<<<nbG, 128, 0, stream>>>(x, W1, tmp, N);
  k_prop_init<<<nbF, 256, 0, stream>>>(tmp, dinv, agg, total);
  k_prop_edges<<<2048, 256, 0, stream>>>(src, dst, ew, dinv, tmp, agg, E);
  k_selu_bias<<<nbF, 256, 0, stream>>>(agg, b1, h, total);

  // layer 2: h = selu(prop(h @ W2) + b2)
  k_gemm_n64<<<nbG, 128, 0, stream>>>(h, W2, tmp, N);
  k_prop_init<<<nbF, 256, 0, stream>>>(tmp, dinv, agg, total);
  k_prop_edges<<<2048, 256, 0, stream>>>(src, dst, ew, dinv, tmp, agg, E);
  k_selu_bias<<<nbF, 256, 0, stream>>>(agg, b2, h, total);

  // assignment + pooling + losses
  k_assign<<<nbG, 128, 0, stream>>>(h, Wp, bp, s_out, N);
  k_pool<<<1024, 256, 0, stream>>>(s_out, h, pool, N);
  k_adj<<<2048, 256, 0, stream>>>(src, dst, ew, s_out, adjA, E);
  k_smoments<<<1024, 256, 0, stream>>>(s_out, dsrc, ssA, ssm, ca, N);
  k_finalize<<<1, 256, 0, stream>>>(pool, adjA, ssA, ssm, ca, twom,
                                    out, N, outOff, adjOff, scalOff);
}